// ShapeBlock_3728031613560
// MI455X (gfx1250) — compile-verified
//
#include <hip/hip_runtime.h>
#include <math.h>

// CDNA5 / gfx1250, wave32. One workgroup (256 thr = 8 waves) per batch row.
// Sliding-window Pearson-correlation scan done with V_WMMA_F32_16X16X4_F32:
//   B (4x16)  = 16 windows (N axis) x 4 k-values      (built from LDS)
//   A (16x4)  = row0: shapelet chunk, row1: ones      (SP, S1 in D rows 0,1)
//   A2(16x4)  = row0: ones, B2 = B*B                  (S2 in D2 row 0)
// A operands are tile-invariant -> hoisted to registers. B operands are
// software-pipelined 2 chunks deep so ds_load latency is covered by the
// WMMA chain instead of an s_wait_dscnt 0 stall every chunk.

typedef __attribute__((ext_vector_type(2))) float v2f;
typedef __attribute__((ext_vector_type(8))) float v8f;

#define KLEN     64
#define SEG_LEN  8064          // END-START = 14064-6000
#define NWIN     8001          // SEG_LEN - K + 1
#define NTILES   504           // 504*16 window slots; 8 waves x 63 tiles
#define LDS_SEG  8160          // padded: max read idx 8126, zero-padded tail
#define E_OUT    32

__global__ __launch_bounds__(256) void shapelet_corr_kernel(
    const float* __restrict__ x,
    const float* __restrict__ shapelet,
    const float* __restrict__ l1_w, const float* __restrict__ l1_b,
    const float* __restrict__ l2_w, const float* __restrict__ l2_b,
    float* __restrict__ out)
{
    __shared__ float seg[LDS_SEG];
    __shared__ float sh[KLEN];
    __shared__ float s_mean_sh;
    __shared__ unsigned long long red;

    const int tid = threadIdx.x;
    const int b   = blockIdx.x;

    // ---- stage global slice -> LDS (coalesced; only 4.1MB of x is touched) ----
    const float* gx = x + ((size_t)(b * 16 + 8) * 40000 + 6000);
    for (int i = tid; i < SEG_LEN; i += 256) seg[i] = gx[i];
    for (int i = SEG_LEN + tid; i < LDS_SEG; i += 256) seg[i] = 0.0f;
    if (tid < KLEN) sh[tid] = shapelet[tid];
    if (tid == 0) red = 0ull;
    __syncthreads();
    if (tid == 0) {
        float s = 0.0f;
        for (int k = 0; k < KLEN; ++k) s += sh[k];
        s_mean_sh = s * (1.0f / KLEN);
    }
    __syncthreads();
    const float s_mean = s_mean_sh;

    const int lane = tid & 31;
    const int wave = tid >> 5;
    const int l    = lane & 15;   // M (for A/D) or N (for B) sub-index
    const int hi   = lane >> 4;   // which K-half this lane carries

    // ---- hoisted A operands (tile-invariant): unconditional LDS reads +
    //      cndmask selects, no exec-mask branching in the hot loop.
    const float onesA = (l == 1) ? 1.0f : 0.0f;
    v2f A1[16];
    #pragma unroll
    for (int c = 0; c < 16; ++c) {
        const float s0 = sh[4 * c + 2 * hi];       // broadcast read, all lanes
        const float s1 = sh[4 * c + 2 * hi + 1];
        A1[c].x = (l == 0) ? s0 : onesA;           // v_cndmask, no branches
        A1[c].y = (l == 0) ? s1 : onesA;
    }
    v2f A2;                                        // row0 = ones
    A2.x = (l == 0) ? 1.0f : 0.0f;
    A2.y = A2.x;

    float bestScore = -INFINITY;
    int   bestW     = 0;

    for (int t = wave; t < NTILES; t += 8) {        // 63 tiles per wave
        const int w0 = t * 16;
        const float* lp = seg + (w0 + l + 2 * hi);  // lane's B base pointer

        // 2-stage pipelined B feed: load chunk c+2 before computing chunk c.
        v2f bc, bn;                                 // current / next B operands
        bc.x = lp[0];  bc.y = lp[1];                // chunk 0
        bn.x = lp[4];  bn.y = lp[5];                // chunk 1

        v8f d1 = {};   // row0: SP, row1: S1
        v8f d2 = {};   // row0: S2
        #pragma unroll
        for (int c = 0; c < 16; ++c) {
            v2f bf = bn;                            // default feed-through
            if (c + 2 < 16) {                       // static under full unroll
                bf.x = lp[4 * (c + 2)];
                bf.y = lp[4 * (c + 2) + 1];
            }
            v2f B2;
            B2.x = bc.x * bc.x;
            B2.y = bc.y * bc.y;
            d1 = __builtin_amdgcn_wmma_f32_16x16x4_f32(false, A1[c], false, bc,
                                                       (short)0, d1, false, false);
            d2 = __builtin_amdgcn_wmma_f32_16x16x4_f32(false, A2, false, B2,
                                                       (short)0, d2, false, false);
            bc = bn;
            bn = bf;
        }

        // lanes 0-15 own window w = w0 + l (D VGPR0 lane n = row0, VGPR1 = row1)
        const int w = w0 + l;
        const float SP = d1[0];
        const float S1 = d1[1];
        const float S2 = d2[0];
        const float cov = SP - S1 * s_mean;                       // scaled cov (argmax-equiv)
        const float var = S2 - S1 * S1 * (1.0f / KLEN);           // scaled variance
        float score = cov * rsqrtf(fmaxf(var, 1e-30f));
        const bool valid = (hi == 0) && (w < NWIN);
        score = valid ? score : -INFINITY;                        // branchless: EXEC stays full
        if (score > bestScore) { bestScore = score; bestW = w; }  // strict > keeps earliest w
    }

    // ---- block argmax: order-preserving float key, ~idx for first-occurrence ties ----
    unsigned int fb  = __float_as_uint(bestScore);
    unsigned int key = (fb & 0x80000000u) ? ~fb : (fb | 0x80000000u);
    unsigned long long pk =
        ((unsigned long long)key << 32) |
        (unsigned long long)(0xFFFFFFFFu - (unsigned int)bestW);
    atomicMax(&red, pk);
    __syncthreads();
    const int bw = (int)(0xFFFFFFFFu - (unsigned int)(red & 0xFFFFFFFFull));

    // ---- fused head: out[b,e] = best.l1_w[e] + l1_b[e] - (sh.l2_w[e] + l2_b[e]) ----
    if (tid < E_OUT) {
        const int e = tid;
        float acc = l1_b[e] - l2_b[e];
        const float* w1 = l1_w + e * KLEN;
        const float* w2 = l2_w + e * KLEN;
        #pragma unroll 8
        for (int k = 0; k < KLEN; ++k)
            acc += seg[bw + k] * w1[k] - sh[k] * w2[k];
        out[b * E_OUT + e] = acc;
    }
}

extern "C" void kernel_launch(void* const* d_in, const int* in_sizes, int n_in,
                              void* d_out, int out_size, void* d_ws, size_t ws_size,
                              hipStream_t stream) {
    (void)in_sizes; (void)n_in; (void)out_size; (void)d_ws; (void)ws_size;
    const float* x        = (const float*)d_in[0];
    const float* shapelet = (const float*)d_in[1];
    const float* l1_w     = (const float*)d_in[2];
    const float* l1_b     = (const float*)d_in[3];
    const float* l2_w     = (const float*)d_in[4];
    const float* l2_b     = (const float*)d_in[5];
    float* out = (float*)d_out;
    shapelet_corr_kernel<<<128, 256, 0, stream>>>(x, shapelet, l1_w, l1_b,
                                                  l2_w, l2_b, out);
}